// VectorQuantizer_1331439861829
// MI455X (gfx1250) — compile-verified
//
#include <hip/hip_runtime.h>

typedef __attribute__((ext_vector_type(2))) float v2f;
typedef __attribute__((ext_vector_type(4))) float v4f;
typedef __attribute__((ext_vector_type(8))) float v8f;

#define VQ_B 64
#define VQ_C 64
#define VQ_HW 1024   // 32*32
#define VQ_K 1024
#define WAVES 8
#define XS_STRIDE 68 // 64 + 4 pad: half-waves hit disjoint LDS bank sets

// ---------------------------------------------------------------------------
// Kernel 1: cn2h[k] = 0.5 * ||codebook[k]||^2   (4 KB into workspace)
// ---------------------------------------------------------------------------
__global__ __launch_bounds__(256) void vq_cnorm_kernel(const float* __restrict__ cb,
                                                       float* __restrict__ cn2h) {
  int k = blockIdx.x * blockDim.x + threadIdx.x;
  if (k < VQ_K) {
    const float* row = cb + (size_t)k * VQ_C;
    float s = 0.0f;
#pragma unroll
    for (int c = 0; c < VQ_C; c += 4) {
      v4f v = *(const v4f*)(row + c);
      s += v.x * v.x + v.y * v.y + v.z * v.z + v.w * v.w;
    }
    cn2h[k] = 0.5f * s;
  }
}

// Preload all 16 B fragments of one 16-code tile into registers (one clause).
__device__ __forceinline__ void vq_ldb(const float* __restrict__ cb, int code, int bk,
                                       v2f (&bf)[16]) {
  const v2f* bp = (const v2f*)cb + (size_t)code * (VQ_C / 2) + bk;
#pragma unroll
  for (int kk = 0; kk < 16; ++kk) bf[kk] = bp[kk * 2];
}

// One 16x16 score tile: two independent WMMA accumulator chains over K=64,
// then fused (dot - 0.5||c||^2) running argmax update.
__device__ __forceinline__ void vq_tile(const v2f (&a)[16], const v2f (&bf)[16],
                                        int code, const float* __restrict__ cn2h,
                                        float (&best)[8], int (&bidx)[8]) {
  const float cn = cn2h[code];
  v8f acc0 = {}, acc1 = {};
#pragma unroll
  for (int kk = 0; kk < 16; kk += 2) {
    acc0 = __builtin_amdgcn_wmma_f32_16x16x4_f32(
        false, a[kk + 0], false, bf[kk + 0], (short)0, acc0, false, false);
    acc1 = __builtin_amdgcn_wmma_f32_16x16x4_f32(
        false, a[kk + 1], false, bf[kk + 1], (short)0, acc1, false, false);
  }
  // D layout: VGPR r, lanes 0-15 -> M=r, lanes 16-31 -> M=r+8; N = lane&15.
  // Codes ascend across tiles, so strict '>' keeps the lowest index on ties.
#pragma unroll
  for (int r = 0; r < 8; ++r) {
    float s = (acc0[r] + acc1[r]) - cn;
    if (s > best[r]) { best[r] = s; bidx[r] = code; }
  }
}

// ---------------------------------------------------------------------------
// Kernel 2: one workgroup (8 waves) per 16-pixel tile.
//   scores(16x1024) = X(16x64) @ CB^T(64x1024) via v_wmma_f32_16x16x4_f32,
//   double-buffered B streaming, fused argmin, gather + NCHW scatter.
// ---------------------------------------------------------------------------
__global__ __launch_bounds__(256) void vq_main_kernel(const float* __restrict__ in,
                                                      const float* __restrict__ cb,
                                                      const float* __restrict__ cn2h,
                                                      float* __restrict__ out) {
  __shared__ float xs[16 * XS_STRIDE];   // staged X tile, [n][c]
  __shared__ float red_s[16 * WAVES];    // per-(row,wave) best score
  __shared__ int   red_i[16 * WAVES];    // per-(row,wave) best index
  __shared__ int   rowbest[16];          // final winning code per row

  const int tid  = threadIdx.x;
  const int lane = tid & 31;
  const int wave = tid >> 5;
  const int m    = blockIdx.x;          // 16-row tile id (0..4095)
  const int b    = m >> 6;              // batch element (HW=1024 => 64 tiles/b)
  const int hw0  = (m & 63) << 4;       // base pixel index within batch elem

  // ---- stage X tile to LDS: xs[n][c] = in[b][c][hw0+n] ----
  {
    const int c  = tid >> 2;            // 0..63
    const int dn = (tid & 3) << 2;      // 0,4,8,12
    v4f v = *(const v4f*)(in + (((size_t)(b * VQ_C + c)) << 10) + hw0 + dn);
    xs[(dn + 0) * XS_STRIDE + c] = v.x;
    xs[(dn + 1) * XS_STRIDE + c] = v.y;
    xs[(dn + 2) * XS_STRIDE + c] = v.z;
    xs[(dn + 3) * XS_STRIDE + c] = v.w;
  }
  __syncthreads();

  // ---- A fragments (fp32 16x4 layout): lane L holds row M=L&15,
  //      K = {0,1} for lanes 0-15, K = {2,3} for lanes 16-31 ----
  const int an = lane & 15;
  const int ak = (lane >> 4) << 1;      // 0 or 2
  v2f a[16];
#pragma unroll
  for (int kk = 0; kk < 16; ++kk)
    a[kk] = *(const v2f*)(xs + an * XS_STRIDE + kk * 4 + ak);

  // ---- running argmax over this wave's 8 codebook N-tiles ----
  float best[8];
  int   bidx[8];
#pragma unroll
  for (int r = 0; r < 8; ++r) { best[r] = -3.0e38f; bidx[r] = 0; }

  const int ncol = lane & 15;           // N position in D tile
  const int bk   = lane >> 4;           // which K-pair this lane holds for B
  const int base = wave * 8;            // first N-tile for this wave

  // ---- double-buffered sweep over 8 codebook tiles ----
  v2f bf0[16], bf1[16];
  vq_ldb(cb, (base + 0) * 16 + ncol, bk, bf0);
#pragma unroll 1
  for (int tt = 0; tt < 4; ++tt) {
    const int t0 = 2 * tt, t1 = 2 * tt + 1;
    vq_ldb(cb, (base + t1) * 16 + ncol, bk, bf1);                 // prefetch odd
    vq_tile(a, bf0, (base + t0) * 16 + ncol, cn2h, best, bidx);   // compute even
    if (tt < 3)
      vq_ldb(cb, (base + t1 + 1) * 16 + ncol, bk, bf0);           // prefetch next even
    vq_tile(a, bf1, (base + t1) * 16 + ncol, cn2h, best, bidx);   // compute odd
  }

  // ---- cross-lane argmax within each 16-lane half (butterfly) ----
#pragma unroll
  for (int r = 0; r < 8; ++r) {
    float s = best[r];
    int   i = bidx[r];
#pragma unroll
    for (int msk = 8; msk >= 1; msk >>= 1) {
      float os = __shfl_xor(s, msk, 32);
      int   oi = __shfl_xor(i, msk, 32);
      if (os > s || (os == s && oi < i)) { s = os; i = oi; }
    }
    const int rowm = r + ((lane >> 4) << 3);   // M = r (lo half) or r+8 (hi half)
    if ((lane & 15) == 0) {
      red_s[rowm * WAVES + wave] = s;
      red_i[rowm * WAVES + wave] = i;
    }
  }
  __syncthreads();

  // ---- cross-wave reduction: 16 threads, 8 candidates each ----
  if (tid < 16) {
    float s = red_s[tid * WAVES];
    int   i = red_i[tid * WAVES];
#pragma unroll
    for (int w = 1; w < WAVES; ++w) {
      float os = red_s[tid * WAVES + w];
      int   oi = red_i[tid * WAVES + w];
      if (os > s || (os == s && oi < i)) { s = os; i = oi; }
    }
    rowbest[tid] = i;
  }
  __syncthreads();

  // ---- gather winning codebook rows, scatter NCHW (coalesced in hw) ----
  {
    const int row = tid & 15;
    const int c0  = tid >> 4;
    const float* crow = cb + (size_t)rowbest[row] * VQ_C;
    const size_t obase = (((size_t)b * VQ_C) << 10) + hw0 + row;
#pragma unroll
    for (int j = 0; j < 4; ++j) {
      const int c = c0 + 16 * j;
      out[obase + ((size_t)c << 10)] = crow[c];
    }
  }
}

// ---------------------------------------------------------------------------
extern "C" void kernel_launch(void* const* d_in, const int* in_sizes, int n_in,
                              void* d_out, int out_size, void* d_ws, size_t ws_size,
                              hipStream_t stream) {
  const float* in  = (const float*)d_in[0];   // [64,64,32,32] f32
  const float* cb  = (const float*)d_in[1];   // [1024,64] f32
  float* out  = (float*)d_out;                // [64,64,32,32] f32
  float* cn2h = (float*)d_ws;                 // 1024 floats scratch

  vq_cnorm_kernel<<<VQ_K / 256, 256, 0, stream>>>(cb, cn2h);
  vq_main_kernel<<<(VQ_B * VQ_HW) / 16, 256, 0, stream>>>(in, cb, cn2h, out);
}